// Z_Layer_88450556494671
// MI455X (gfx1250) — compile-verified
//
#include <hip/hip_runtime.h>
#include <math.h>

// Problem sizes (fixed by the reference)
#define B_   64
#define T_   2048
#define NIN_ 256
#define H_   64
#define G4_  256   // 4*H

typedef float v2f __attribute__((ext_vector_type(2)));
typedef float v8f __attribute__((ext_vector_type(8)));
typedef unsigned int v4u __attribute__((ext_vector_type(4)));
typedef int v4i __attribute__((ext_vector_type(4)));
typedef int v8i __attribute__((ext_vector_type(8)));

__device__ __forceinline__ float sigm(float x) { return 1.0f / (1.0f + __expf(-x)); }

// LDS row pitch for the staged A tile: 256 data DWORDs + 2 pad DWORDs
// (inserted by TDM pad_interval=256dw / pad_amount=2dw) -> bank-conflict-free
// ds_load_b64 across lanes (row*258 mod 64 banks == 2*row).
#define APITCH 258

// ---------------------------------------------------------------------------
// TDM: issue tensor_load_to_lds for a 16x256 f32 tile whose rows are strided
// by T_*NIN_ elements in global memory. Descriptor per CDNA5 ISA ch.8.
// ---------------------------------------------------------------------------
#if __has_builtin(__builtin_amdgcn_tensor_load_to_lds)
#define HAVE_TDM 1
__device__ __forceinline__ void tdm_load_tile(unsigned int lds_off, const float* gptr) {
  unsigned long long ga = (unsigned long long)(uintptr_t)gptr;
  v4u g0;
  g0.x = 1u;                                        // count=1, user mode
  g0.y = lds_off;                                   // lds_addr (bytes)
  g0.z = (unsigned int)ga;                          // global_addr[31:0]
  g0.w = (unsigned int)((ga >> 32) & 0x01FFFFFFu)   // global_addr[56:32]
         | 0x80000000u;                             // type=2 ("image")
  v8i g1;
  g1[0] = (int)((2u << 16)      // data_size = 4 bytes
              | (1u << 20)      // pad_enable
              | (7u << 22)      // pad_interval: 256 DWORDs
              | (1u << 25));    // pad_amount: 2 DWORDs
  g1[1] = (int)(256u << 16);    // tensor_dim0 = 256 (low16 -> bits[63:48])
  g1[2] = (int)(16u << 16);     // tensor_dim1 = 16  (low16 -> bits[95:80])
  g1[3] = (int)(256u << 16);    // tile_dim0 = 256   (bits[127:112])
  g1[4] = 16;                   // tile_dim1 = 16    (bits[143:128]); tile_dim2=0
  g1[5] = (int)(T_ * NIN_);     // tensor_dim0_stride = 524288 elements
  g1[6] = 0;                    // dim0_stride hi / dim1_stride lo
  g1[7] = 0;
  v4i z4 = {0, 0, 0, 0};
#if __clang_major__ >= 23
  v8i z8 = {0, 0, 0, 0, 0, 0, 0, 0};
  __builtin_amdgcn_tensor_load_to_lds(g0, g1, z4, z4, z8, 0);
#else
  __builtin_amdgcn_tensor_load_to_lds(g0, g1, z4, z4, 0);
#endif
}
#else
#define HAVE_TDM 0
#endif

// ---------------------------------------------------------------------------
// Kernel 1: time-parallel input projection via f32 WMMA.
//   Xg[t][b][j] = sum_k X[b][t][k] * W_ih[j][k] + b_ih[j] + b_hh[j]
// One block = one (timestep, 16-row batch tile); TDM stages the 16x256 A tile
// into LDS once, then 8 waves compute two 16x16 output tiles each.
// ---------------------------------------------------------------------------
__global__ __launch_bounds__(256) void xg_wmma_kernel(
    const float* __restrict__ X,      // [B, T, NIN]
    const float* __restrict__ Wih,    // [G4, NIN+1]
    const float* __restrict__ bih,    // [G4]
    const float* __restrict__ bhh,    // [G4]
    float* __restrict__ Xg)           // [T, B, G4]
{
  __shared__ float Xs[16 * APITCH];   // 16.5 KB staged A tile (padded pitch)

  const int lane = threadIdx.x & 31;
  const int wv   = threadIdx.x >> 5;
  const int tt   = blockIdx.x >> 2;   // timestep
  const int mb   = blockIdx.x & 3;    // batch 16-row tile
  const int mr   = lane & 15;
  const int kh   = lane >> 4;         // which K-half this lane holds

  const float* tile_base = X + ((size_t)(mb * 16) * T_ + tt) * NIN_;

#if HAVE_TDM
  if (threadIdx.x == 0)
    tdm_load_tile((unsigned int)(uintptr_t)&Xs[0], tile_base);
  __builtin_amdgcn_s_wait_tensorcnt(0);
#else
  for (int i = threadIdx.x; i < 16 * 256; i += 256) {
    const int r = i >> 8, c = i & 255;
    Xs[r * APITCH + c] = tile_base[(size_t)r * T_ * NIN_ + c];
  }
#endif
  __syncthreads();

  const int n0 = wv * 32 + mr;
  const int n1 = n0 + 16;
  const float* w0 = Wih + (size_t)n0 * (NIN_ + 1);
  const float* w1 = Wih + (size_t)n1 * (NIN_ + 1);
  const float* arow = &Xs[mr * APITCH];

  v8f acc0 = {0.f, 0.f, 0.f, 0.f, 0.f, 0.f, 0.f, 0.f};
  v8f acc1 = acc0;

  for (int kk = 0; kk < NIN_ / 4; ++kk) {
    const int kb = kk * 4 + kh * 2;
    v2f a;  a.x  = arow[kb]; a.y  = arow[kb + 1];   // ds_load_b64
    v2f b0; b0.x = w0[kb];   b0.y = w0[kb + 1];
    v2f b1; b1.x = w1[kb];   b1.y = w1[kb + 1];
    acc0 = __builtin_amdgcn_wmma_f32_16x16x4_f32(false, a, false, b0, (short)0, acc0, false, false);
    acc1 = __builtin_amdgcn_wmma_f32_16x16x4_f32(false, a, false, b1, (short)0, acc1, false, false);
  }

  float* ob = Xg + (size_t)tt * B_ * G4_;
#pragma unroll
  for (int r = 0; r < 8; ++r) {
    const int row = mb * 16 + kh * 8 + r;
    ob[(size_t)row * G4_ + n0] = acc0[r] + bih[n0] + bhh[n0];
    ob[(size_t)row * G4_ + n1] = acc1[r] + bih[n1] + bhh[n1];
  }
}

// ---------------------------------------------------------------------------
// Kernel 1b: PX[t*64+b] = bias2 + X[b][t][:] . w21   (time-parallel matvec)
// ---------------------------------------------------------------------------
__global__ __launch_bounds__(256) void px_kernel(
    const float* __restrict__ X, const float* __restrict__ w21,
    const float* __restrict__ bias2, float* __restrict__ PX)
{
  const int i = blockIdx.x * 256 + threadIdx.x;   // i = t*64 + b, exact cover
  const int t = i >> 6, b = i & 63;
  const float* xr = X + ((size_t)b * T_ + t) * NIN_;
  float s = bias2[0];
#pragma unroll 8
  for (int k = 0; k < NIN_; ++k) s = fmaf(xr[k], w21[k], s);
  PX[i] = s;
}

// ---------------------------------------------------------------------------
// Kernel 2: recurrent scan. One workgroup per batch element (64 independent
// LSTM chains). Thread j owns gate column j: W_hh row j and w22 live in VGPRs,
// h is broadcast from LDS each step. Two barriers per step.
// ---------------------------------------------------------------------------
__global__ __launch_bounds__(256) void scan_kernel(
    const float* __restrict__ Xg,    // [T, B, G4]
    const float* __restrict__ PX,    // [T*B]
    const float* __restrict__ Whh,   // [G4, H]
    const float* __restrict__ Wih,   // [G4, NIN+1] (last column only)
    const float* __restrict__ w22,   // [H]
    float* __restrict__ Z)           // [B, T]
{
  __shared__ float hsh[H_];
  __shared__ float gsh[G4_];

  const int tid = threadIdx.x;
  const int b   = blockIdx.x;

  float Wreg[H_], w22reg[H_];
#pragma unroll
  for (int k = 0; k < H_; ++k) Wreg[k] = Whh[(size_t)tid * H_ + k];
#pragma unroll
  for (int k = 0; k < H_; ++k) w22reg[k] = w22[k];
  const float wlast = Wih[(size_t)tid * (NIN_ + 1) + NIN_];

  if (tid < H_) hsh[tid] = 0.0f;
  float creg = 0.0f;
  __syncthreads();

  for (int t = 0; t < T_; ++t) {
    const float* xgt = Xg + ((size_t)t * B_ + b) * G4_;
    if (tid < 4)  // pull next step's 1KB gate slab toward the WGP
      __builtin_prefetch(xgt + (size_t)B_ * G4_ + tid * 64, 0, 0);

    float hvec[H_];
#pragma unroll
    for (int k = 0; k < H_; ++k) hvec[k] = hsh[k];   // LDS broadcast reads

    // pz = sigmoid(px + h . w22)  (computed redundantly; uniform per block)
    float s = PX[(size_t)t * B_ + b];
#pragma unroll
    for (int k = 0; k < H_; ++k) s = fmaf(hvec[k], w22reg[k], s);
    const float pz = sigm(s);

    // gate column j: Xg + pz*W_ih[j,256] + h . W_hh[j,:]
    float g = xgt[tid] + pz * wlast;
#pragma unroll
    for (int k = 0; k < H_; ++k) g = fmaf(hvec[k], Wreg[k], g);
    gsh[tid] = g;
    __syncthreads();

    if (tid < H_) {
      const float ig = sigm(gsh[tid]);
      const float fg = sigm(gsh[H_ + tid]);
      const float gg = tanhf(gsh[2 * H_ + tid]);
      const float og = sigm(gsh[3 * H_ + tid]);
      creg = fmaf(fg, creg, ig * gg);
      hsh[tid] = og * tanhf(creg);
    }
    if (tid == 0) Z[(size_t)b * T_ + t] = pz;
    __syncthreads();
  }
}

// ---------------------------------------------------------------------------
// Launch: workspace = Xg (T*B*G4 f32 = 128 MiB) + PX (T*B f32 = 512 KiB).
// ---------------------------------------------------------------------------
extern "C" void kernel_launch(void* const* d_in, const int* in_sizes, int n_in,
                              void* d_out, int out_size, void* d_ws, size_t ws_size,
                              hipStream_t stream) {
  (void)in_sizes; (void)n_in; (void)out_size; (void)ws_size;
  const float* X     = (const float*)d_in[0];  // h_concat [64,2048,256]
  const float* w21   = (const float*)d_in[1];  // [256,1]
  const float* w22   = (const float*)d_in[2];  // [64,1]
  const float* bias2 = (const float*)d_in[3];  // [1]
  const float* Wih   = (const float*)d_in[4];  // [256,257]
  const float* Whh   = (const float*)d_in[5];  // [256,64]
  const float* bih   = (const float*)d_in[6];  // [256]
  const float* bhh   = (const float*)d_in[7];  // [256]

  float* Z  = (float*)d_out;                   // [64,2048]
  float* Xg = (float*)d_ws;                    // [T,B,G4]
  float* PX = Xg + (size_t)T_ * B_ * G4_;      // [T*B]

  xg_wmma_kernel<<<dim3(T_ * (B_ / 16)), dim3(256), 0, stream>>>(X, Wih, bih, bhh, Xg);
  px_kernel<<<dim3((B_ * T_) / 256), dim3(256), 0, stream>>>(X, w21, bias2, PX);
  scan_kernel<<<dim3(B_), dim3(256), 0, stream>>>(Xg, PX, Whh, Wih, w22, Z);
}